// CosCelHead_34583076667969
// MI455X (gfx1250) — compile-verified
//
#include <hip/hip_runtime.h>
#include <stdint.h>

// Problem constants (match reference)
#define B_ROWS 512
#define C_COLS 65536
#define BLOCK  256                 // 8 waves of 32
#define NWAVES (BLOCK / 32)

// Streaming tiling for pass 1
#define CHUNKS 4                                  // blocks per row
#define COLS_PER_BLOCK (C_COLS / CHUNKS)          // 16384 floats
#define TILE 2048                                 // floats per array per tile
#define TILE_BYTES (TILE * 4)                     // 8192 B
#define ARR_BYTES  TILE_BYTES                     // one array's tile in LDS
#define BUF_BYTES  (2 * ARR_BYTES)                // pred+targ per buffer
#define NTILES (COLS_PER_BLOCK / TILE)            // 8

typedef __attribute__((ext_vector_type(2))) float v2f;
typedef __attribute__((ext_vector_type(8))) float v8f;

// ---------------------------------------------------------------------------
// Exact wave32 sum using V_WMMA_F32_16X16X4_F32.
//   A[m,0]=p_m (lanes 0-15, VGPR0), A[m,2]=p_{m+16} (lanes 16-31, VGPR0),
//   A[m,1]=A[m,3]=0 (VGPR1). B = all ones (layout independent).
//   => D[m,n] = p_m + p_{m+16} for every n.
//   D VGPR j holds rows j (lanes 0-15) and j+8 (lanes 16-31), so summing the
//   8 D registers gives lanes 0-15: sum_{m=0..7}, lanes 16-31: sum_{m=8..15};
//   one xor-16 shuffle finishes the reduction (total in all lanes).
// EXEC must be all ones at the call site (uniform control flow).
// ---------------------------------------------------------------------------
__device__ __forceinline__ float wave_sum_wmma(float p) {
  v2f a   = {p, 0.0f};
  v2f one = {1.0f, 1.0f};
  v8f c   = {};
  v8f d = __builtin_amdgcn_wmma_f32_16x16x4_f32(
      /*neg_a=*/false, a, /*neg_b=*/false, one,
      /*c_mod=*/(short)0, c, /*reuse_a=*/false, /*reuse_b=*/false);
  float s = ((d[0] + d[1]) + (d[2] + d[3])) + ((d[4] + d[5]) + (d[6] + d[7]));
  s += __shfl_xor(s, 16, 32);
  return s;
}

// Issue one tile (pred + targ) of this thread's slots into an LDS buffer via
// the CDNA5 async-to-LDS path. Each thread owns float4 slots {tid, tid+256};
// the instruction's IOFFSET is added to BOTH the LDS and global addresses, so
// a single offset:4096 covers the second slot in both spaces.
__device__ __forceinline__ void issue_async_tile(const float* predC,
                                                 const float* targC,
                                                 uint32_t ldsBase, int buf,
                                                 int tile, int tid) {
  const uint64_t gP = (uint64_t)(uintptr_t)predC +
                      (uint64_t)tile * TILE_BYTES + (uint64_t)(tid * 16);
  const uint64_t gT = (uint64_t)(uintptr_t)targC +
                      (uint64_t)tile * TILE_BYTES + (uint64_t)(tid * 16);
  const uint32_t lP = ldsBase + (uint32_t)(buf * BUF_BYTES) + (uint32_t)(tid * 16);
  const uint32_t lT = lP + ARR_BYTES;
  asm volatile(
      "global_load_async_to_lds_b128 %0, %2, off\n\t"
      "global_load_async_to_lds_b128 %0, %2, off offset:4096\n\t"
      "global_load_async_to_lds_b128 %1, %3, off\n\t"
      "global_load_async_to_lds_b128 %1, %3, off offset:4096"
      :
      : "v"(lP), "v"(lT), "v"(gP), "v"(gT)
      : "memory");
}

// ---------------------------------------------------------------------------
// Pass 1: each block reduces one (row, chunk) strip of 16384 columns into
// float4{ sum(a*b), sum(a*a), sum(b*b), sum(bce*bce_w) } where
// a = targ*cw, b = pred*cw,  bce = max(t,0) - t*p + log1p(exp(-|t|)).
// Double-buffered async streaming: each lane consumes only the LDS bytes it
// loaded itself, so only s_wait_asynccnt (no barriers) gates the stream loop.
// ---------------------------------------------------------------------------
__global__ __launch_bounds__(BLOCK) void cce_pass1(
    const float* __restrict__ pred, const float* __restrict__ targ,
    const float* __restrict__ cw, const float* __restrict__ bw,
    float4* __restrict__ partial) {
  __shared__ __align__(16) float ldsbuf[2][2][TILE];   // [buf][pred/targ][elems]
  __shared__ float sred[NWAVES][4];

  const int tid  = threadIdx.x;
  const int part = blockIdx.x & (CHUNKS - 1);
  const int row  = blockIdx.x >> 2;
  const size_t rowBase = (size_t)row * C_COLS + (size_t)part * COLS_PER_BLOCK;
  const float* predC = pred + rowBase;
  const float* targC = targ + rowBase;
  const int colBase4 = (part * COLS_PER_BLOCK) >> 2;   // float4 index into weights

  const uint32_t ldsBase = (uint32_t)(uintptr_t)(&ldsbuf[0][0][0]);
  const float4* cw4 = (const float4*)cw;
  const float4* gw4 = (const float4*)bw;

  float num = 0.0f, na2 = 0.0f, nb2 = 0.0f, bces = 0.0f;

  issue_async_tile(predC, targC, ldsBase, 0, 0, tid);  // prime the pipeline
  int buf = 0;
  for (int tile = 0; tile < NTILES; ++tile) {
    if (tile + 1 < NTILES) {
      issue_async_tile(predC, targC, ldsBase, buf ^ 1, tile + 1, tid);
      // 4 newer async ops may stay in flight; wait for current tile's 4.
      asm volatile("s_wait_asynccnt 0x4" ::: "memory");
    } else {
      asm volatile("s_wait_asynccnt 0x0" ::: "memory");
    }
    const float4* lp4 = (const float4*)(&ldsbuf[buf][0][0]);
    const float4* lt4 = (const float4*)(&ldsbuf[buf][1][0]);
#pragma unroll
    for (int j = 0; j < 2; ++j) {
      const int v  = tid + j * BLOCK;              // float4 slot in tile
      const float4 p = lp4[v];
      const float4 t = lt4[v];
      const int c4 = colBase4 + tile * (TILE / 4) + v;
      const float4 w = cw4[c4];
      const float4 g = gw4[c4];
#define ACC(pv, tv, wv, gv)                                           \
      {                                                               \
        const float aa = (tv) * (wv);                                 \
        const float bb = (pv) * (wv);                                 \
        num += aa * bb;                                               \
        na2 += aa * aa;                                               \
        nb2 += bb * bb;                                               \
        const float bce = fmaxf((tv), 0.0f) - (tv) * (pv) +           \
                          log1pf(expf(-fabsf((tv))));                 \
        bces += bce * (gv);                                           \
      }
      ACC(p.x, t.x, w.x, g.x)
      ACC(p.y, t.y, w.y, g.y)
      ACC(p.z, t.z, w.z, g.z)
      ACC(p.w, t.w, w.w, g.w)
#undef ACC
    }
    buf ^= 1;
  }

  // Wave-level exact sums via WMMA, then cross-wave via LDS.
  num  = wave_sum_wmma(num);
  na2  = wave_sum_wmma(na2);
  nb2  = wave_sum_wmma(nb2);
  bces = wave_sum_wmma(bces);

  const int lane = tid & 31, wv = tid >> 5;
  if (lane == 0) {
    sred[wv][0] = num; sred[wv][1] = na2; sred[wv][2] = nb2; sred[wv][3] = bces;
  }
  __syncthreads();
  if (tid == 0) {
    float4 o = {0.0f, 0.0f, 0.0f, 0.0f};
    for (int w = 0; w < NWAVES; ++w) {
      o.x += sred[w][0]; o.y += sred[w][1];
      o.z += sred[w][2]; o.w += sred[w][3];
    }
    partial[blockIdx.x] = o;
  }
}

// ---------------------------------------------------------------------------
// Pass 2 (single block): fold chunk partials per row into cosine terms,
// reduce bce weights over C, combine into the scalar loss.
// ---------------------------------------------------------------------------
__global__ __launch_bounds__(BLOCK) void cce_pass2(
    const float* __restrict__ bw, const float4* __restrict__ partial,
    float* __restrict__ out) {
  __shared__ float sred[NWAVES][3];
  const int tid = threadIdx.x;

  const float4* bw4 = (const float4*)bw;
  float wsum = 0.0f;
  for (int i = tid; i < C_COLS / 4; i += BLOCK) {
    const float4 b = bw4[i];
    wsum += (b.x + b.y) + (b.z + b.w);
  }

  float cos_sum = 0.0f, bce_sum = 0.0f;
  for (int r = tid; r < B_ROWS; r += BLOCK) {
    float n = 0.0f, x2 = 0.0f, y2 = 0.0f, e = 0.0f;
#pragma unroll
    for (int k = 0; k < CHUNKS; ++k) {
      const float4 v = partial[r * CHUNKS + k];
      n += v.x; x2 += v.y; y2 += v.z; e += v.w;
    }
    const float na = fmaxf(sqrtf(x2), 1e-8f);
    const float nb = fmaxf(sqrtf(y2), 1e-8f);
    cos_sum += n / (na * nb);
    bce_sum += e;
  }

  wsum    = wave_sum_wmma(wsum);
  cos_sum = wave_sum_wmma(cos_sum);
  bce_sum = wave_sum_wmma(bce_sum);

  const int lane = tid & 31, wv = tid >> 5;
  if (lane == 0) {
    sred[wv][0] = wsum; sred[wv][1] = cos_sum; sred[wv][2] = bce_sum;
  }
  __syncthreads();
  if (tid == 0) {
    float W = 0.0f, Cs = 0.0f, E = 0.0f;
    for (int w = 0; w < NWAVES; ++w) {
      W += sred[w][0]; Cs += sred[w][1]; E += sred[w][2];
    }
    out[0] = -(Cs / (float)B_ROWS) + (E / (W + 1e-10f)) * 0.1f;
  }
}

extern "C" void kernel_launch(void* const* d_in, const int* in_sizes, int n_in,
                              void* d_out, int out_size, void* d_ws,
                              size_t ws_size, hipStream_t stream) {
  (void)in_sizes; (void)n_in; (void)out_size; (void)ws_size;
  const float* pred = (const float*)d_in[0];
  const float* targ = (const float*)d_in[1];
  const float* cw   = (const float*)d_in[2];
  const float* bw   = (const float*)d_in[3];
  float4* partial = (float4*)d_ws;   // B_ROWS*CHUNKS float4 = 32 KB

  cce_pass1<<<B_ROWS * CHUNKS, BLOCK, 0, stream>>>(pred, targ, cw, bw, partial);
  cce_pass2<<<1, BLOCK, 0, stream>>>(bw, partial, (float*)d_out);
}